// LennardJonesPotential_13400297963575
// MI455X (gfx1250) — compile-verified
//
#include <hip/hip_runtime.h>

typedef __attribute__((ext_vector_type(2))) float v2f;
typedef __attribute__((ext_vector_type(8))) float v8f;

#define RCUT2 6.25f   // (2.5 * sigma)^2, sigma = 1

__global__ void lj_zero_kernel(float* out) { out[0] = 0.0f; }

__global__ __launch_bounds__(256) void lj_wmma_kernel(const float* __restrict__ pos,
                                                      float* __restrict__ out,
                                                      int N) {
    __shared__ float  sraw[3072];   // raw packed (x,y,z), async-DMA target (12 KB)
    __shared__ float4 sB[1024];     // (x, y, z, |x|^2) per particle, B-fragment layout (16 KB)

    const int tid = threadIdx.x;
    const int jchunk = blockIdx.y * 1024;

    // Triangular scheduling: whole block strictly below the diagonal -> no i<j pairs.
    // Uniform exit before any barrier.
    if (blockIdx.x * 128 >= jchunk + 1024) return;

    // ---- stage 1: CDNA5 async global->LDS DMA of the raw j-chunk (ASYNCcnt) ----
    {
        const int nj = (N - jchunk) < 1024 ? (N - jchunk) : 1024;
        const int total_dw = 3 * nj;
        const float* gbase = pos + (size_t)jchunk * 3;
        #pragma unroll
        for (int k = 0; k < 12; ++k) {
            const int idx = k * 256 + tid;
            if (idx < total_dw) {
                unsigned lds_addr = (unsigned)(uintptr_t)&sraw[idx];
                const float* g = gbase + idx;
                asm volatile("global_load_async_to_lds_b32 %0, %1, off"
                             :: "v"(lds_addr), "v"(g)
                             : "memory");
            } else {
                sraw[idx] = 1.0e18f;   // pad -> huge finite r^2 -> fails cutoff mask
            }
        }
        asm volatile("s_wait_asynccnt 0" ::: "memory");
    }
    __syncthreads();

    // ---- stage 2: transform to B-fragment layout (x, y, z, |x|^2) ----
    #pragma unroll
    for (int k = 0; k < 4; ++k) {
        const int p = k * 256 + tid;
        const float x = sraw[3 * p + 0];
        const float y = sraw[3 * p + 1];
        const float z = sraw[3 * p + 2];
        sB[p] = make_float4(x, y, z, x * x + y * y + z * z);
    }
    __syncthreads();

    // Scalarize the per-wave coordinates: keeps all tile-loop control and the
    // diagonal guard in SALU, so EXEC is untouched around every WMMA.
    const int wave = __builtin_amdgcn_readfirstlane(tid >> 5);
    const int lane = tid & 31;
    const bool hi  = lane >= 16;
    const int  m   = lane & 15;

    const int ibase = blockIdx.x * 128 + wave * 16;   // scalar
    const int irow  = ibase + m;

    float ax, ay, az;
    if (irow < N) {
        ax = pos[3 * irow + 0];
        ay = pos[3 * irow + 1];
        az = pos[3 * irow + 2];
    } else {
        ax = -1.0e18f; ay = -1.0e18f; az = -1.0e18f;
    }
    const float an2 = ax * ax + ay * ay + az * az;

    // A fragment (16x4 f32), row i = (-2x_i, -2y_i, -2z_i, 1):
    //   lanes 0-15:  (K0,K1) = (-2x_i, -2y_i);  lanes 16-31: (K2,K3) = (-2z_i, 1)
    v2f a;
    a.x = hi ? (-2.0f * az) : (-2.0f * ax);
    a.y = hi ? 1.0f         : (-2.0f * ay);

    // C_ij = |x_i|^2 (row-only): c[v] = an2 of row (v + 8*hi). Loop-invariant,
    // built once with 8 lane shuffles (ds_bpermute).
    v8f c;
    #pragma unroll
    for (int v = 0; v < 8; ++v)
        c[v] = __shfl(an2, v + (hi ? 8 : 0), 32);

    // B fragment source: lanes 0-15 read (x,y), lanes 16-31 read (z,|x|^2)
    const float2* sB2 = (const float2*)sB;
    const int col2base = 2 * m + (hi ? 1 : 0);

    float acc = 0.0f;

    // Off-diagonal tile body: every element already has i<j; only cutoff mask.
    auto tile = [&](int tt) {
        const float2 qb = sB2[tt * 32 + col2base];
        v2f b; b.x = qb.x; b.y = qb.y;
        v8f r = __builtin_amdgcn_wmma_f32_16x16x4_f32(
            false, a, false, b, (short)0, c, false, false);
        #pragma unroll
        for (int v = 0; v < 8; ++v) {
            const bool keep = r[v] < RCUT2;           // NaN/inf -> false
            const float sr2 = __builtin_amdgcn_rcpf(keep ? r[v] : 1.0f);
            const float sr6 = sr2 * sr2 * sr2;
            acc += 4.0f * (sr6 * sr6 - sr6);          // masked pairs give exactly 0
        }
    };

    const int d = ibase - jchunk;   // scalar, multiple of 16; >=0 iff diagonal here
    int t = 0;

    if (d >= 0) {
        // Diagonal tile: jbase == ibase, needs the strict i<j integer mask.
        t = d >> 4;
        const float2 qb = sB2[t * 32 + col2base];
        v2f b; b.x = qb.x; b.y = qb.y;
        v8f r = __builtin_amdgcn_wmma_f32_16x16x4_f32(
            false, a, false, b, (short)0, c, false, false);
        const int thr = m - (hi ? 8 : 0);   // keep rows with v < thr
        #pragma unroll
        for (int v = 0; v < 8; ++v) {
            const bool keep = (v < thr) && (r[v] < RCUT2);
            const float sr2 = __builtin_amdgcn_rcpf(keep ? r[v] : 1.0f);
            const float sr6 = sr2 * sr2 * sr2;
            acc += 4.0f * (sr6 * sr6 - sr6);
        }
        ++t;
    }

    // Manually jammed 4-tile groups: 4 independent WMMA + VALU chains per
    // iteration for ILP and amortized loop overhead; scalar trip count.
    for (; t + 3 < 64; t += 4) {
        tile(t);
        tile(t + 1);
        tile(t + 2);
        tile(t + 3);
    }
    for (; t < 64; ++t)
        tile(t);

    // wave32 tree reduction, then one global f32 atomic per wave
    #pragma unroll
    for (int off = 16; off >= 1; off >>= 1)
        acc += __shfl_xor(acc, off, 32);

    if (lane == 0)
        atomicAdd(out, acc);
}

extern "C" void kernel_launch(void* const* d_in, const int* in_sizes, int n_in,
                              void* d_out, int out_size, void* d_ws, size_t ws_size,
                              hipStream_t stream) {
    const float* pos = (const float*)d_in[0];
    float* out = (float*)d_out;
    const int N = in_sizes[0] / 3;   // positions is [N, 3] f32

    lj_zero_kernel<<<1, 1, 0, stream>>>(out);

    dim3 grid((N + 127) / 128,      // 8 i-tiles (of 16 rows) per block
              (N + 1023) / 1024);   // 1024-particle j-chunks
    lj_wmma_kernel<<<grid, 256, 0, stream>>>(pos, out, N);
}